// GCNBlock_19980187861403
// MI455X (gfx1250) — compile-verified
//
#include <hip/hip_runtime.h>

typedef float v2f __attribute__((ext_vector_type(2)));
typedef float v8f __attribute__((ext_vector_type(8)));

#define H 128
#define NTILES 8  // 128 / 16 column tiles

// ---------------------------------------------------------------- utilities
__device__ __forceinline__ void atomAddF(float* p, float v) {
#if defined(__HIP_PLATFORM_AMD__)
    unsafeAtomicAdd(p, v);   // hardware global_atomic_add_f32
#else
    atomicAdd(p, v);
#endif
}

// ---------------------------------------------------------------- degree
__global__ void k_zero_deg(float* __restrict__ deg, int n) {
    int i = blockIdx.x * 256 + threadIdx.x;
    if (i < n) deg[i] = 0.0f;
}

__global__ void k_count_deg(const long long* __restrict__ ei,
                            float* __restrict__ deg, int E) {
    int e = blockIdx.x * 256 + threadIdx.x;
    if (e < E) {
        int d = (int)ei[(size_t)E + e];  // dst row
        atomAddF(&deg[d], 1.0f);
    }
}

__global__ void k_dinv(float* __restrict__ deg, int n) {
    int i = blockIdx.x * 256 + threadIdx.x;
    if (i < n) deg[i] = rsqrtf(deg[i] + 2.0f);  // improved=True self-loop wt 2
}

// ---------------------------------------------------------------- weight pack
// P[n][k/2] = { Wc[k][n], Wc[k+1][n], Ws[k][n], Ws[k+1][n] }
// -> one global_load_b128 per B fragment pair (both GEMMs) in the main loop.
__global__ void k_pack(const float* __restrict__ Wc, const float* __restrict__ Wsk,
                       float4* __restrict__ P) {
    int i = blockIdx.x * 256 + threadIdx.x;   // 128*64 = 8192 entries
    int n  = i >> 6;
    int kp = i & 63;
    int k  = kp * 2;
    float4 v;
    v.x = Wc[(size_t)k * H + n];
    v.y = Wc[(size_t)(k + 1) * H + n];
    v.z = Wsk[(size_t)k * H + n];
    v.w = Wsk[(size_t)(k + 1) * H + n];
    P[i] = v;
}

// ---------------------------------------------------------------- fused dual GEMM
// h = x @ W_conv  (stored to workspace)
// out = 2*dinv^2 * h + x @ W_skip + b_conv + b_skip   (baseline before scatter)
// One wave computes 16 rows x 128 cols via V_WMMA_F32_16X16X4_F32.
__global__ __launch_bounds__(256) void k_gemm(
    const float* __restrict__ x,
    const float4* __restrict__ P,
    const float* __restrict__ bc, const float* __restrict__ bsk,
    const float* __restrict__ dinv,
    float* __restrict__ h, float* __restrict__ out, int nnodes)
{
    const int lane = threadIdx.x & 31;
    const int wave = threadIdx.x >> 5;
    const int rowstart = blockIdx.x * 128 + wave * 16;

    const int mloc  = lane & 15;          // A row / B,C,D column within tile
    const int khalf = (lane >> 4) * 2;    // lanes 0-15 -> K 0,1 ; lanes 16-31 -> K 2,3

    int mrow = rowstart + mloc;
    if (mrow >= nnodes) mrow = nnodes - 1;          // clamp: keep EXEC full for WMMA
    const float* xrow = x + (size_t)mrow * H;

    // per-tile packed-weight row pointers (n = t*16 + mloc)
    const float4* Pn[NTILES];
#pragma unroll
    for (int t = 0; t < NTILES; ++t)
        Pn[t] = P + (size_t)(t * 16 + mloc) * (H / 2);

    v8f zero{};
    v8f accC[NTILES], accS[NTILES];
#pragma unroll
    for (int t = 0; t < NTILES; ++t) { accC[t] = zero; accS[t] = zero; }

#pragma unroll 4
    for (int k0 = 0; k0 < H; k0 += 4) {
        v2f a;
        a.x = xrow[k0 + khalf];
        a.y = xrow[k0 + khalf + 1];
        const int kp = (k0 >> 1) + (lane >> 4);     // packed-K index
#pragma unroll
        for (int t = 0; t < NTILES; ++t) {
            const float4 pv = Pn[t][kp];            // b128: both B fragments
            v2f b, c;
            b.x = pv.x; b.y = pv.y;                 // W_conv frag
            c.x = pv.z; c.y = pv.w;                 // W_skip frag
            accC[t] = __builtin_amdgcn_wmma_f32_16x16x4_f32(
                false, a, false, b, (short)0, accC[t], false, false);
            accS[t] = __builtin_amdgcn_wmma_f32_16x16x4_f32(
                false, a, false, c, (short)0, accS[t], false, false);
        }
    }

    // C/D layout: VGPR v -> row v (lanes 0-15) or v+8 (lanes 16-31), col = lane&15
    const int rofs = (lane >> 4) * 8;
#pragma unroll
    for (int v = 0; v < 8; ++v) {
        const int gm = rowstart + rofs + v;
        if (gm < nnodes) {
            const float di = dinv[gm];
            const float sl = 2.0f * di * di;       // self-loop weight * norm
            const size_t rb = (size_t)gm * H;
#pragma unroll
            for (int t = 0; t < NTILES; ++t) {
                const int n = t * 16 + mloc;
                const float hv = accC[t][v];
                const float sv = accS[t][v];
                h[rb + n]   = hv;
                out[rb + n] = sl * hv + sv + bc[n] + bsk[n];
            }
        }
    }
}

// ---------------------------------------------------------------- edge scatter
// out[dst] += dinv[src]*dinv[dst] * h[src]; one wave per edge, float4 per lane.
__global__ __launch_bounds__(256) void k_scatter(
    const long long* __restrict__ ei, const float* __restrict__ h,
    const float* __restrict__ dinv, float* __restrict__ out, int E)
{
    const int lane = threadIdx.x & 31;
    const int e = blockIdx.x * 8 + (threadIdx.x >> 5);
    if (e >= E) return;
    const int s = (int)ei[e];
    const int d = (int)ei[(size_t)E + e];
    const float w = dinv[s] * dinv[d];
    const float4 hv = ((const float4*)(h + (size_t)s * H))[lane];
    float* orow = out + (size_t)d * H + lane * 4;
    atomAddF(orow + 0, w * hv.x);
    atomAddF(orow + 1, w * hv.y);
    atomAddF(orow + 2, w * hv.z);
    atomAddF(orow + 3, w * hv.w);
}

// ---------------------------------------------------------------- ELU epilogue
__global__ void k_elu(float* __restrict__ out, int n) {
    int i = blockIdx.x * 256 + threadIdx.x;
    if (i < n) {
        const float v = out[i];
        out[i] = v > 0.0f ? v : 0.1f * (expf(v) - 1.0f);
    }
}

// ---------------------------------------------------------------- launcher
extern "C" void kernel_launch(void* const* d_in, const int* in_sizes, int n_in,
                              void* d_out, int out_size, void* d_ws, size_t ws_size,
                              hipStream_t stream) {
    const float*     x   = (const float*)d_in[0];
    const long long* ei  = (const long long*)d_in[1];   // int64 [2, E]
    const float*     Wc  = (const float*)d_in[2];
    const float*     bc  = (const float*)d_in[3];
    const float*     Wsk = (const float*)d_in[4];
    const float*     bsk = (const float*)d_in[5];
    float*           out = (float*)d_out;

    const int nn = in_sizes[0] / H;   // 100000 nodes
    const int E  = in_sizes[1] / 2;   // 1600000 edges

    // workspace layout: deg/dinv | packed weights P | h
    float*  deg = (float*)d_ws;
    char*   p0  = (char*)d_ws + (((size_t)nn * sizeof(float) + 255) & ~(size_t)255);
    float4* P   = (float4*)p0;                               // 128*64 float4 = 128KB
    float*  h   = (float*)(p0 + (size_t)H * (H / 2) * sizeof(float4));

    const int nb_n = (nn + 255) / 256;
    k_zero_deg<<<nb_n, 256, 0, stream>>>(deg, nn);
    k_count_deg<<<(E + 255) / 256, 256, 0, stream>>>(ei, deg, E);
    k_dinv<<<nb_n, 256, 0, stream>>>(deg, nn);

    k_pack<<<(H * (H / 2) + 255) / 256, 256, 0, stream>>>(Wc, Wsk, P);

    k_gemm<<<(nn + 127) / 128, 256, 0, stream>>>(x, P, bc, bsk, deg, h, out, nn);

    k_scatter<<<(E + 7) / 8, 256, 0, stream>>>(ei, h, deg, out, E);

    const int tot = nn * H;
    k_elu<<<(tot + 255) / 256, 256, 0, stream>>>(out, tot);
}